// GNN_34376918237435
// MI455X (gfx1250) — compile-verified
//
#include <hip/hip_runtime.h>

// ---------------------------------------------------------------------------
// GNN forward on MI455X (gfx1250, wave32).
//  - All GEMMs via V_WMMA_F32_16X16X4_F32 (fp32-exact, one wave per 16x16 tile)
//  - Edge aggregation via float4 gathers + global_atomic_add_f32 (L2-resident)
// ---------------------------------------------------------------------------

typedef __attribute__((ext_vector_type(2))) float v2f;
typedef __attribute__((ext_vector_type(8))) float v8f;

// out[M,Cout] = act( A[M,Cin] @ W1 + (DUAL ? B[M,Cin] @ W2 : 0) + bias )
// One wave computes one 16x16 tile, stepping K by 4 with f32 WMMA.
// Requires: M % 16 == 0, Cout % 16 == 0, Cin % 4 == 0 (all true here).
template <bool RELU, bool DUAL>
__global__ __launch_bounds__(256) void gemm_wmma_kernel(
    const float* __restrict__ A, const float* __restrict__ W1,
    const float* __restrict__ B, const float* __restrict__ W2,
    const float* __restrict__ bias, float* __restrict__ out,
    int M, int Cin, int Cout)
{
    const int lane = threadIdx.x & 31;
    const int wavesPerBlock = blockDim.x >> 5;
    const int tile = blockIdx.x * wavesPerBlock + (threadIdx.x >> 5);
    const int ntn = Cout >> 4;
    const int ntm = M >> 4;
    if (tile >= ntm * ntn) return;          // whole-wave exit: EXEC stays all-1s
    const int tm = tile / ntn;
    const int tn = tile - tm * ntn;

    const int l    = lane & 15;
    const int koff = (lane >> 4) << 1;      // lanes 0-15 -> K+0,K+1 ; 16-31 -> K+2,K+3
    const int col  = tn * 16 + l;

    const float* arow = A + (size_t)(tm * 16 + l) * Cin;
    const float* brow = DUAL ? (B + (size_t)(tm * 16 + l) * Cin) : nullptr;

    v8f acc = {};
    for (int k = 0; k < Cin; k += 4) {
        v2f av; av.x = arow[k + koff];
                av.y = arow[k + koff + 1];
        v2f wv; wv.x = W1[(size_t)(k + koff)     * Cout + col];
                wv.y = W1[(size_t)(k + koff + 1) * Cout + col];
        acc = __builtin_amdgcn_wmma_f32_16x16x4_f32(
                  false, av, false, wv, (short)0, acc, false, false);
        if (DUAL) {
            v2f bv; bv.x = brow[k + koff];
                    bv.y = brow[k + koff + 1];
            v2f w2; w2.x = W2[(size_t)(k + koff)     * Cout + col];
                    w2.y = W2[(size_t)(k + koff + 1) * Cout + col];
            acc = __builtin_amdgcn_wmma_f32_16x16x4_f32(
                      false, bv, false, w2, (short)0, acc, false, false);
        }
    }

    // C/D layout: VGPR r -> row (tm*16 + r) for lanes 0-15, (tm*16 + 8 + r) for 16-31
    const float bval = bias[col];
    const int m0 = tm * 16 + ((lane >> 4) << 3);
    #pragma unroll
    for (int r = 0; r < 8; ++r) {
        float v = acc[r] + bval;
        if (RELU) v = fmaxf(v, 0.0f);
        out[(size_t)(m0 + r) * Cout + col] = v;
    }
}

// agg[dst[e]*C + c..c+3] += h[src[e]*C + c..c+3]
// One thread per (edge, 4-channel group): one b128 gather + 4 f32 atomics.
__global__ __launch_bounds__(256) void scatter_add_kernel(
    const float* __restrict__ h, const int* __restrict__ src,
    const int* __restrict__ dst, float* __restrict__ agg,
    long long E, int cshift)   // channels = 1 << cshift (64 or 128)
{
    long long idx = (long long)blockIdx.x * blockDim.x + threadIdx.x;
    const int g4shift = cshift - 2;               // float4 groups per edge
    const long long total = E << g4shift;
    if (idx >= total) return;
    const int e  = (int)(idx >> g4shift);
    const int c4 = (int)(idx & ((1 << g4shift) - 1)) << 2;
    const int s  = src[e];
    const int d  = dst[e];
    const float4 v = *(const float4*)(h + ((size_t)s << cshift) + c4);
    float* dp = agg + ((size_t)d << cshift) + c4;
    atomicAdd(dp + 0, v.x);
    atomicAdd(dp + 1, v.y);
    atomicAdd(dp + 2, v.z);
    atomicAdd(dp + 3, v.w);
}

__global__ __launch_bounds__(256) void zero_kernel(float4* __restrict__ p, long long n4)
{
    long long i = (long long)blockIdx.x * blockDim.x + threadIdx.x;
    if (i < n4) p[i] = make_float4(0.f, 0.f, 0.f, 0.f);
}

// sums[batch[n]*128 + c..c+3] += h[n*128 + c..c+3];  cnt[batch[n]] += 1 per node
__global__ __launch_bounds__(256) void pool_sum_kernel(
    const float* __restrict__ h, const int* __restrict__ batch,
    float* __restrict__ sums, float* __restrict__ cnt, long long N)
{
    long long idx = (long long)blockIdx.x * blockDim.x + threadIdx.x;
    const long long total = N << 5;               // 128/4 = 32 float4 groups/node
    if (idx >= total) return;
    const int node = (int)(idx >> 5);
    const int c4   = (int)(idx & 31) << 2;
    const int g    = batch[node];
    const float4 v = *(const float4*)(h + ((size_t)node << 7) + c4);
    float* sp = sums + ((size_t)g << 7) + c4;
    atomicAdd(sp + 0, v.x);
    atomicAdd(sp + 1, v.y);
    atomicAdd(sp + 2, v.z);
    atomicAdd(sp + 3, v.w);
    if (c4 == 0) atomicAdd(&cnt[g], 1.0f);
}

__global__ __launch_bounds__(256) void pool_div_kernel(
    const float4* __restrict__ sums, const float* __restrict__ cnt,
    float4* __restrict__ pooled, int G)
{
    int idx = blockIdx.x * blockDim.x + threadIdx.x;
    if (idx >= (G << 5)) return;                  // G * 32 float4 groups
    const float m = fmaxf(cnt[idx >> 5], 1.0f);
    float4 s = sums[idx];
    pooled[idx] = make_float4(s.x / m, s.y / m, s.z / m, s.w / m);
}

// ---------------------------------------------------------------------------

static inline unsigned blocks_for(long long n, int t) {
    return (unsigned)((n + t - 1) / t);
}

extern "C" void kernel_launch(void* const* d_in, const int* in_sizes, int n_in,
                              void* d_out, int out_size, void* d_ws, size_t ws_size,
                              hipStream_t stream)
{
    const float* x        = (const float*)d_in[0];
    const int*   edge     = (const int*)  d_in[1];
    const int*   batch    = (const int*)  d_in[2];
    // d_in[3] = num_graphs (derived from out_size instead)
    const float* w_stage1 = (const float*)d_in[4];
    const float* b_stage1 = (const float*)d_in[5];
    const float* rel1_w   = (const float*)d_in[6];
    const float* rel1_b   = (const float*)d_in[7];
    const float* root1_w  = (const float*)d_in[8];
    const float* rel2_w   = (const float*)d_in[9];
    const float* rel2_b   = (const float*)d_in[10];
    const float* root2_w  = (const float*)d_in[11];
    const float* rel3_w   = (const float*)d_in[12];
    const float* rel3_b   = (const float*)d_in[13];
    const float* root3_w  = (const float*)d_in[14];
    const float* lin_w    = (const float*)d_in[15];
    const float* lin_b    = (const float*)d_in[16];
    float*       out      = (float*)d_out;

    const int N = in_sizes[0] / 128;   // 50000
    const int E = in_sizes[1] / 2;     // 800000
    const int G = out_size / 16;       // 512
    const int* src = edge;
    const int* dst = edge + E;

    float* ws = (float*)d_ws;
    size_t off = 0;
    float* bufA   = ws + off; off += (size_t)N * 128;
    float* bufB   = ws + off; off += (size_t)N * 128;
    float* agg    = ws + off; off += (size_t)N * 128;
    float* sums   = ws + off; off += (size_t)G * 128;
    float* cnt    = ws + off; off += (size_t)G;
    float* pooled = ws + off; off += (size_t)G * 128;
    (void)ws_size; (void)n_in;

    const int T = 256;
    const int WPB = T / 32;  // waves (16x16 tiles) per block

    auto gemm_grid = [&](int M, int Cout) {
        return blocks_for((long long)(M >> 4) * (Cout >> 4), WPB);
    };

    // ---- stage 1: h = x @ w_stage1 + b                    [N,128] -> [N,64]
    gemm_wmma_kernel<false, false><<<gemm_grid(N, 64), T, 0, stream>>>(
        x, w_stage1, nullptr, nullptr, b_stage1, bufA, N, 128, 64);

    // ---- conv layer 1 (64 -> 64), ReLU
    zero_kernel<<<blocks_for((long long)N * 64 / 4, T), T, 0, stream>>>(
        (float4*)agg, (long long)N * 64 / 4);
    scatter_add_kernel<<<blocks_for((long long)E << 4, T), T, 0, stream>>>(
        bufA, src, dst, agg, E, 6);
    gemm_wmma_kernel<true, true><<<gemm_grid(N, 64), T, 0, stream>>>(
        agg, rel1_w, bufA, root1_w, rel1_b, bufB, N, 64, 64);

    // ---- conv layer 2 (64 -> 128), ReLU
    zero_kernel<<<blocks_for((long long)N * 64 / 4, T), T, 0, stream>>>(
        (float4*)agg, (long long)N * 64 / 4);
    scatter_add_kernel<<<blocks_for((long long)E << 4, T), T, 0, stream>>>(
        bufB, src, dst, agg, E, 6);
    gemm_wmma_kernel<true, true><<<gemm_grid(N, 128), T, 0, stream>>>(
        agg, rel2_w, bufB, root2_w, rel2_b, bufA, N, 64, 128);

    // ---- conv layer 3 (128 -> 128), ReLU
    zero_kernel<<<blocks_for((long long)N * 128 / 4, T), T, 0, stream>>>(
        (float4*)agg, (long long)N * 128 / 4);
    scatter_add_kernel<<<blocks_for((long long)E << 5, T), T, 0, stream>>>(
        bufA, src, dst, agg, E, 7);
    gemm_wmma_kernel<true, true><<<gemm_grid(N, 128), T, 0, stream>>>(
        agg, rel3_w, bufA, root3_w, rel3_b, bufB, N, 128, 128);

    // ---- global mean pool over graphs
    zero_kernel<<<blocks_for((long long)G * 128 / 4, T), T, 0, stream>>>(
        (float4*)sums, (long long)G * 128 / 4);
    zero_kernel<<<blocks_for((long long)G / 4, T), T, 0, stream>>>(
        (float4*)cnt, (long long)G / 4);
    pool_sum_kernel<<<blocks_for((long long)N << 5, T), T, 0, stream>>>(
        bufB, batch, sums, cnt, N);
    pool_div_kernel<<<blocks_for((long long)G << 5, T), T, 0, stream>>>(
        (const float4*)sums, cnt, (float4*)pooled, G);

    // ---- classifier: out = pooled @ lin_w + lin_b          [G,128] -> [G,16]
    gemm_wmma_kernel<false, false><<<gemm_grid(G, 16), T, 0, stream>>>(
        pooled, lin_w, nullptr, nullptr, lin_b, out, G, 128, 16);
}